// NTM_13846974562458
// MI455X (gfx1250) — compile-verified
//
#include <hip/hip_runtime.h>
#include <hip/hip_bf16.h>
#include <math.h>

#define EPSV 1e-6f
constexpr int BB  = 64;     // batch
constexpr int VV  = 10000;  // vocab
constexpr int EE  = 256;    // embed
constexpr int HH  = 512;    // hidden
constexpr int GG  = 2048;   // 4*H gates
constexpr int NN  = 8192;   // memory slots (power of 2)
constexpr int FF  = 128;    // feature
constexpr int OWN = 390;    // write interp out
constexpr int ORN = 134;    // read interp out
constexpr int CH  = 16;     // rvec chunks

typedef __attribute__((ext_vector_type(2))) float v2f;
typedef __attribute__((ext_vector_type(8))) float v8f;

__device__ __forceinline__ float sigf(float x)  { return 1.f / (1.f + __expf(-x)); }
__device__ __forceinline__ float splus(float x) { return x > 20.f ? x : log1pf(__expf(x)); }

// ---------------------------------------------------------------------------
// Gates GEMM: gates[b,j] = sum_k A[b,k]*W[j,k] + b_ih[j] + b_hh[j]
// A[b,k] = (k<E) ? emb[x[b],k] : h[b,k-E].  M=64, N=2048, K=768, fp32 WMMA.
// ---------------------------------------------------------------------------
__global__ void k_gates_wmma(const int* __restrict__ x, const float* __restrict__ emb,
                             const float* __restrict__ h,
                             const float* __restrict__ W_ih, const float* __restrict__ b_ih,
                             const float* __restrict__ W_hh, const float* __restrict__ b_hh,
                             float* __restrict__ gates) {
    const int tilesN = GG / 16;                      // 128
    int wid  = (blockIdx.x * blockDim.x + threadIdx.x) >> 5;
    int lane = threadIdx.x & 31;
    int tm = wid / tilesN, tn = wid % tilesN;
    if (tm >= BB / 16) return;
    int half = lane >> 4, l16 = lane & 15;
    int m = tm * 16 + l16;
    int n = tn * 16 + l16;
    const float* Ae = emb + (size_t)x[m] * EE;       // K in [0,256)
    const float* Ah = h   + (size_t)m * HH;          // K-256
    const float* Wi = W_ih + (size_t)n * EE;
    const float* Wh = W_hh + (size_t)n * HH;
    v8f acc = {};
    for (int k0 = 0; k0 < EE; k0 += 4) {
        int k = k0 + 2 * half;
        v2f a; a.x = Ae[k]; a.y = Ae[k + 1];
        v2f b; b.x = Wi[k]; b.y = Wi[k + 1];
        acc = __builtin_amdgcn_wmma_f32_16x16x4_f32(false, a, false, b, (short)0, acc, false, false);
    }
    for (int k0 = 0; k0 < HH; k0 += 4) {
        int k = k0 + 2 * half;
        v2f a; a.x = Ah[k]; a.y = Ah[k + 1];
        v2f b; b.x = Wh[k]; b.y = Wh[k + 1];
        acc = __builtin_amdgcn_wmma_f32_16x16x4_f32(false, a, false, b, (short)0, acc, false, false);
    }
    for (int i = 0; i < 8; ++i) {
        int row = tm * 16 + half * 8 + i;
        int col = tn * 16 + l16;
        gates[(size_t)row * GG + col] = acc[i] + b_ih[col] + b_hh[col];
    }
}

// ---------------------------------------------------------------------------
// Generic fp32 WMMA GEMM: C[M,N] = A[M,K] * W[N,K]^T + bias[N]  (K % 4 == 0)
// ---------------------------------------------------------------------------
__global__ void k_gemm_nt_wmma(const float* __restrict__ A, int lda,
                               const float* __restrict__ W, int ldw,
                               const float* __restrict__ bias,
                               float* __restrict__ C, int ldc,
                               int M, int N, int K) {
    int tilesN = (N + 15) >> 4;
    int tilesM = (M + 15) >> 4;
    int wid  = (blockIdx.x * blockDim.x + threadIdx.x) >> 5;
    int lane = threadIdx.x & 31;
    if (wid >= tilesM * tilesN) return;
    int tm = wid / tilesN, tn = wid % tilesN;
    int half = lane >> 4, l16 = lane & 15;
    int m = tm * 16 + l16;
    int n = tn * 16 + l16;
    bool mv = m < M, nv = n < N;
    const float* Ap = A + (size_t)(mv ? m : 0) * lda;
    const float* Wp = W + (size_t)(nv ? n : 0) * ldw;
    v8f acc = {};
    for (int k0 = 0; k0 < K; k0 += 4) {
        int k = k0 + 2 * half;
        v2f a; a.x = mv ? Ap[k] : 0.f; a.y = mv ? Ap[k + 1] : 0.f;
        v2f b; b.x = nv ? Wp[k] : 0.f; b.y = nv ? Wp[k + 1] : 0.f;
        acc = __builtin_amdgcn_wmma_f32_16x16x4_f32(false, a, false, b, (short)0, acc, false, false);
    }
    for (int i = 0; i < 8; ++i) {
        int row = tm * 16 + half * 8 + i;
        int col = tn * 16 + l16;
        if (row < M && col < N)
            C[(size_t)row * ldc + col] = acc[i] + (bias ? bias[col] : 0.f);
    }
}

// ---------------------------------------------------------------------------
// LSTM pointwise: h2 = sig(o)*tanh(sig(f)*c + sig(i)*tanh(g))
// ---------------------------------------------------------------------------
__global__ void k_lstm(const float* __restrict__ gates, const float* __restrict__ c,
                       float* __restrict__ h2) {
    int idx = blockIdx.x * blockDim.x + threadIdx.x;
    if (idx >= BB * HH) return;
    int b = idx / HH, j = idx % HH;
    const float* g = gates + (size_t)b * GG;
    float gi = g[j], gf = g[HH + j], gg = g[2 * HH + j], go = g[3 * HH + j];
    float c2 = sigf(gf) * c[idx] + sigf(gi) * tanhf(gg);
    h2[idx] = sigf(go) * tanhf(c2);
}

// ---------------------------------------------------------------------------
// Interpret head params (write: k,beta,g,s,r,e,a; read: k,beta,g,s,r) + ||k+EPS||
// ---------------------------------------------------------------------------
__global__ void k_interp(const float* __restrict__ ow, const float* __restrict__ orr,
                         float* kw, float* ew, float* aw, float* kr,
                         float* betaw, float* gw, float* sw, float* rw, float* knw,
                         float* betar, float* gr, float* sr, float* rr, float* knr) {
    __shared__ float red[FF];
    int b = blockIdx.x, t = threadIdx.x;                 // 128 threads
    const float* o = ow  + (size_t)b * OWN;
    const float* p = orr + (size_t)b * ORN;
    float kv = o[t];
    kw[b * FF + t] = kv;
    ew[b * FF + t] = sigf(o[FF + 6 + t]);                // e = sig(o[134:262])
    aw[b * FF + t] = o[2 * FF + 6 + t];                  // a = o[262:390]
    float kv2 = p[t];
    kr[b * FF + t] = kv2;

    red[t] = (kv + EPSV) * (kv + EPSV);
    __syncthreads();
    for (int off = 64; off > 0; off >>= 1) { if (t < off) red[t] += red[t + off]; __syncthreads(); }
    if (t == 0) knw[b] = sqrtf(red[0]);
    __syncthreads();
    red[t] = (kv2 + EPSV) * (kv2 + EPSV);
    __syncthreads();
    for (int off = 64; off > 0; off >>= 1) { if (t < off) red[t] += red[t + off]; __syncthreads(); }
    if (t == 0) knr[b] = sqrtf(red[0]);

    if (t == 0) {
        betaw[b] = splus(o[FF]);
        gw[b]    = sigf(o[FF + 1]);
        float s0 = o[FF + 2], s1 = o[FF + 3], s2 = o[FF + 4];
        float mx = fmaxf(s0, fmaxf(s1, s2));
        float e0 = __expf(s0 - mx), e1 = __expf(s1 - mx), e2 = __expf(s2 - mx);
        float inv = 1.f / (e0 + e1 + e2);
        sw[b * 3 + 0] = e0 * inv; sw[b * 3 + 1] = e1 * inv; sw[b * 3 + 2] = e2 * inv;
        rw[b] = 1.f + splus(o[FF + 5]);
    }
    if (t == 1) {
        betar[b] = splus(p[FF]);
        gr[b]    = sigf(p[FF + 1]);
        float s0 = p[FF + 2], s1 = p[FF + 3], s2 = p[FF + 4];
        float mx = fmaxf(s0, fmaxf(s1, s2));
        float e0 = __expf(s0 - mx), e1 = __expf(s1 - mx), e2 = __expf(s2 - mx);
        float inv = 1.f / (e0 + e1 + e2);
        sr[b * 3 + 0] = e0 * inv; sr[b * 3 + 1] = e1 * inv; sr[b * 3 + 2] = e2 * inv;
        rr[b] = 1.f + splus(p[FF + 5]);
    }
}

// ---------------------------------------------------------------------------
// Pass 1: z[b,n] = beta * cos_sim(bank[b,n]+EPS, k+EPS).  One wave per row.
// ---------------------------------------------------------------------------
__global__ void k_sim_w(const float* __restrict__ bank, const float* __restrict__ kw,
                        const float* __restrict__ knw, const float* __restrict__ betaw,
                        float* __restrict__ z) {
    int b = blockIdx.y;
    int wave = threadIdx.x >> 5, lane = threadIdx.x & 31;
    size_t n = (size_t)blockIdx.x * 8 + wave;
    const float4* row = (const float4*)(bank + ((size_t)b * NN + n) * FF);
    const float4* kp  = (const float4*)(kw + (size_t)b * FF);
    float4 v = row[lane], kk = kp[lane];
    float m0 = v.x + EPSV, m1 = v.y + EPSV, m2 = v.z + EPSV, m3 = v.w + EPSV;
    float d  = m0 * (kk.x + EPSV) + m1 * (kk.y + EPSV) + m2 * (kk.z + EPSV) + m3 * (kk.w + EPSV);
    float sq = m0 * m0 + m1 * m1 + m2 * m2 + m3 * m3;
    for (int off = 16; off > 0; off >>= 1) {
        d  += __shfl_xor(d, off, 32);
        sq += __shfl_xor(sq, off, 32);
    }
    if (lane == 0) {
        float sim = d / fmaxf(sqrtf(sq) * knw[b], 1e-8f);
        z[(size_t)b * NN + n] = betaw[b] * sim;
    }
}

// ---------------------------------------------------------------------------
// Pass 2: same but on bank2 computed on the fly (bank2 never materialized).
// ---------------------------------------------------------------------------
__global__ void k_sim_r(const float* __restrict__ bank, const float* __restrict__ ww,
                        const float* __restrict__ ew, const float* __restrict__ aw,
                        const float* __restrict__ kr, const float* __restrict__ knr,
                        const float* __restrict__ betar, float* __restrict__ z) {
    int b = blockIdx.y;
    int wave = threadIdx.x >> 5, lane = threadIdx.x & 31;
    size_t n = (size_t)blockIdx.x * 8 + wave;
    float wwn = ww[(size_t)b * NN + n];
    const float4* row = (const float4*)(bank + ((size_t)b * NN + n) * FF);
    const float4* ep  = (const float4*)(ew + (size_t)b * FF);
    const float4* ap  = (const float4*)(aw + (size_t)b * FF);
    const float4* kp  = (const float4*)(kr + (size_t)b * FF);
    float4 v = row[lane], e4 = ep[lane], a4 = ap[lane], kk = kp[lane];
    float m0 = v.x * (1.f - wwn * e4.x) + wwn * a4.x + EPSV;
    float m1 = v.y * (1.f - wwn * e4.y) + wwn * a4.y + EPSV;
    float m2 = v.z * (1.f - wwn * e4.z) + wwn * a4.z + EPSV;
    float m3 = v.w * (1.f - wwn * e4.w) + wwn * a4.w + EPSV;
    float d  = m0 * (kk.x + EPSV) + m1 * (kk.y + EPSV) + m2 * (kk.z + EPSV) + m3 * (kk.w + EPSV);
    float sq = m0 * m0 + m1 * m1 + m2 * m2 + m3 * m3;
    for (int off = 16; off > 0; off >>= 1) {
        d  += __shfl_xor(d, off, 32);
        sq += __shfl_xor(sq, off, 32);
    }
    if (lane == 0) {
        float sim = d / fmaxf(sqrtf(sq) * knr[b], 1e-8f);
        z[(size_t)b * NN + n] = betar[b] * sim;
    }
}

// ---------------------------------------------------------------------------
// Head finalize: softmax(z) + (1-g)*w_prev -> circular 3-tap shift -> w^r/sum.
// One block per batch; 8192-entry weight vector staged in LDS (320KB/WGP).
// ---------------------------------------------------------------------------
__global__ void __launch_bounds__(1024)
k_head(const float* __restrict__ z, const float* __restrict__ wprev,
       const float* __restrict__ g_, const float* __restrict__ s_,
       const float* __restrict__ r_, float* __restrict__ wout) {
    __shared__ float wbuf[NN];
    __shared__ float red[1024];
    int b = blockIdx.x, t = threadIdx.x;
    const float* zb = z + (size_t)b * NN;

    float zl[8], mx = -1e30f;
    for (int i = 0; i < 8; ++i) { zl[i] = zb[t + i * 1024]; mx = fmaxf(mx, zl[i]); }
    red[t] = mx; __syncthreads();
    for (int o = 512; o > 0; o >>= 1) { if (t < o) red[t] = fmaxf(red[t], red[t + o]); __syncthreads(); }
    mx = red[0]; __syncthreads();

    float ex[8], sum = 0.f;
    for (int i = 0; i < 8; ++i) { ex[i] = __expf(zl[i] - mx); sum += ex[i]; }
    red[t] = sum; __syncthreads();
    for (int o = 512; o > 0; o >>= 1) { if (t < o) red[t] += red[t + o]; __syncthreads(); }
    float inv = 1.f / red[0]; __syncthreads();

    float gb = g_[b];
    for (int i = 0; i < 8; ++i) {
        int n = t + i * 1024;
        wbuf[n] = ex[i] * inv + (1.f - gb) * wprev[(size_t)b * NN + n];
    }
    __syncthreads();

    float s0 = s_[b * 3 + 0], s1 = s_[b * 3 + 1], s2 = s_[b * 3 + 2];
    float rb = r_[b];
    float p[8], psum = 0.f;
    for (int i = 0; i < 8; ++i) {
        int n = t + i * 1024;
        int nm = (n + NN - 1) & (NN - 1), np = (n + 1) & (NN - 1);
        float w2 = s0 * wbuf[nm] + s1 * wbuf[n] + s2 * wbuf[np];
        p[i] = __powf(w2, rb);
        psum += p[i];
    }
    red[t] = psum; __syncthreads();
    for (int o = 512; o > 0; o >>= 1) { if (t < o) red[t] += red[t + o]; __syncthreads(); }
    float invp = 1.f / (red[0] + EPSV);
    for (int i = 0; i < 8; ++i)
        wout[(size_t)b * NN + t + i * 1024] = p[i] * invp;
}

// ---------------------------------------------------------------------------
// Pass 3: rvec partials. rvec[b,f] = sum_n wr[b,n]*bank2[b,n,f] (bank2 on the fly).
// Deterministic two-stage reduction (no float atomics).
// ---------------------------------------------------------------------------
__global__ void __launch_bounds__(1024)
k_rvec_part(const float* __restrict__ bank, const float* __restrict__ ww,
            const float* __restrict__ wr, const float* __restrict__ ew,
            const float* __restrict__ aw, float* __restrict__ partial) {
    __shared__ float red[1024];
    int b = blockIdx.y, ch = blockIdx.x;
    int t = threadIdx.x;
    int f = t & 127, sl = t >> 7;                    // 8 n-slices x 128 features
    float ef = ew[b * FF + f], af = aw[b * FF + f];
    float acc = 0.f;
    int n0 = ch * (NN / CH);
    for (int n = n0 + sl; n < n0 + NN / CH; n += 8) {
        float wwn = ww[(size_t)b * NN + n];
        float wrn = wr[(size_t)b * NN + n];
        float m2 = bank[((size_t)b * NN + n) * FF + f] * (1.f - wwn * ef) + wwn * af;
        acc += wrn * m2;
    }
    red[t] = acc; __syncthreads();
    for (int o = 512; o >= 128; o >>= 1) { if (t < o) red[t] += red[t + o]; __syncthreads(); }
    if (t < 128) partial[((size_t)b * CH + ch) * FF + t] = red[t];
}

__global__ void k_rvec_red(const float* __restrict__ partial, float* __restrict__ rvec) {
    int b = blockIdx.x, f = threadIdx.x;
    float acc = 0.f;
    for (int ch = 0; ch < CH; ++ch) acc += partial[((size_t)b * CH + ch) * FF + f];
    rvec[b * FF + f] = acc;
}

// ---------------------------------------------------------------------------
extern "C" void kernel_launch(void* const* d_in, const int* in_sizes, int n_in,
                              void* d_out, int out_size, void* d_ws, size_t ws_size,
                              hipStream_t stream) {
    const int*   x       = (const int*)  d_in[0];
    const float* h       = (const float*)d_in[1];
    const float* c       = (const float*)d_in[2];
    const float* w_read  = (const float*)d_in[3];
    const float* w_write = (const float*)d_in[4];
    const float* bank    = (const float*)d_in[5];
    const float* emb     = (const float*)d_in[6];
    const float* W_ih    = (const float*)d_in[7];
    const float* b_ih    = (const float*)d_in[8];
    const float* W_hh    = (const float*)d_in[9];
    const float* b_hh    = (const float*)d_in[10];
    const float* W_rd    = (const float*)d_in[11];
    const float* b_rd    = (const float*)d_in[12];
    const float* W_wr    = (const float*)d_in[13];
    const float* b_wr    = (const float*)d_in[14];
    const float* W_out   = (const float*)d_in[15];
    const float* b_out   = (const float*)d_in[16];
    float* out = (float*)d_out;

    float* ws = (float*)d_ws;
    float* gates = ws;            ws += BB * GG;
    float* h2    = ws;            ws += BB * HH;
    float* ow    = ws;            ws += BB * OWN;
    float* orr   = ws;            ws += BB * ORN;
    float* kw    = ws;            ws += BB * FF;
    float* ew    = ws;            ws += BB * FF;
    float* aw    = ws;            ws += BB * FF;
    float* kr    = ws;            ws += BB * FF;
    float* betaw = ws;            ws += BB;
    float* gw    = ws;            ws += BB;
    float* rw    = ws;            ws += BB;
    float* knw   = ws;            ws += BB;
    float* sw    = ws;            ws += 3 * BB;
    float* betar = ws;            ws += BB;
    float* gr    = ws;            ws += BB;
    float* rr    = ws;            ws += BB;
    float* knr   = ws;            ws += BB;
    float* sr    = ws;            ws += 3 * BB;
    float* zbuf  = ws;            ws += (size_t)BB * NN;   // reused for both heads
    float* wwv   = ws;            ws += (size_t)BB * NN;
    float* wrv   = ws;            ws += (size_t)BB * NN;
    float* part  = ws;            ws += BB * CH * FF;
    float* rvec  = ws;            ws += BB * FF;

    auto gemm = [&](const float* A, int lda, const float* W, int ldw, const float* bias,
                    float* C, int ldc, int M, int N, int K) {
        int tiles = ((M + 15) / 16) * ((N + 15) / 16);
        int blocks = (tiles + 3) / 4;                       // 4 waves / block
        k_gemm_nt_wmma<<<blocks, 128, 0, stream>>>(A, lda, W, ldw, bias, C, ldc, M, N, K);
    };

    // 1) Embedding + gate GEMM (WMMA f32), then LSTM pointwise -> h2
    k_gates_wmma<<<(4 * (GG / 16)) / 4, 128, 0, stream>>>(x, emb, h, W_ih, b_ih, W_hh, b_hh, gates);
    k_lstm<<<(BB * HH + 255) / 256, 256, 0, stream>>>(gates, c, h2);

    // 2) Interpreter GEMMs + param extraction
    gemm(h2, HH, W_wr, HH, b_wr, ow, OWN, BB, OWN, HH);
    gemm(h2, HH, W_rd, HH, b_rd, orr, ORN, BB, ORN, HH);
    k_interp<<<BB, FF, 0, stream>>>(ow, orr, kw, ew, aw, kr,
                                    betaw, gw, sw, rw, knw, betar, gr, sr, rr, knr);

    // 3) Write head: sim pass over bank (256MB), then softmax/shift/sharpen
    k_sim_w<<<dim3(NN / 8, BB), 256, 0, stream>>>(bank, kw, knw, betaw, zbuf);
    k_head<<<BB, 1024, 0, stream>>>(zbuf, w_write, gw, sw, rw, wwv);

    // 4) Read head on bank2 (computed on the fly; bank2 never materialized)
    k_sim_r<<<dim3(NN / 8, BB), 256, 0, stream>>>(bank, wwv, ew, aw, kr, knr, betar, zbuf);
    k_head<<<BB, 1024, 0, stream>>>(zbuf, w_read, gr, sr, rr, wrv);

    // 5) rvec = sum_n wr * bank2 (third and final bank pass), then output GEMM
    k_rvec_part<<<dim3(CH, BB), 1024, 0, stream>>>(bank, wwv, wrv, ew, aw, part);
    k_rvec_red<<<BB, FF, 0, stream>>>(part, rvec);
    gemm(rvec, FF, W_out, FF, b_out, out, VV, BB, VV, FF);
}